// Vanilla_27324581937611
// MI455X (gfx1250) — compile-verified
//
#include <hip/hip_runtime.h>
#include <hip/hip_bf16.h>

// ---------------------------------------------------------------------------
// GCN forward on MI455X (gfx1250). GEMMs use V_WMMA_F32_16X16X4_F32 (fp32
// matrix pipe). Gather/scatter stays in L2 (feature matrix = 25.6MB << 192MB).
// W is staged in LDS pre-packed in K-pairs so each B fragment is one aligned
// ds_load_b64 into an even VGPR pair (no repack movs); A fragments are
// software-pipelined so the global load overlaps the 8-WMMA chain.
// ---------------------------------------------------------------------------

typedef float v2f __attribute__((ext_vector_type(2)));
typedef float v8f __attribute__((ext_vector_type(8)));

#define HID   128
#define ODIM  10

// ---------------------------------------------------------------- prep kernels
__global__ void k_fill(float* p, float v, int n) {
    int i = blockIdx.x * blockDim.x + threadIdx.x;
    if (i < n) p[i] = v;
}

__global__ void k_count_deg(const int* __restrict__ dst, float* deg, int e) {
    int i = blockIdx.x * blockDim.x + threadIdx.x;
    if (i < e) atomicAdd(&deg[dst[i]], 1.0f);
}

__global__ void k_dinv(float* deg_dinv, float* self_norm, int n) {
    int i = blockIdx.x * blockDim.x + threadIdx.x;
    if (i < n) {
        float d = rsqrtf(deg_dinv[i]);
        deg_dinv[i]  = d;       // becomes dinv in place
        self_norm[i] = d * d;
    }
}

__global__ void k_edge_norm(const int* __restrict__ src, const int* __restrict__ dst,
                            const float* __restrict__ dinv, float* en, int e) {
    int i = blockIdx.x * blockDim.x + threadIdx.x;
    if (i < e) en[i] = dinv[src[i]] * dinv[dst[i]];
}

// ------------------------------------------------------------ WMMA fp32 GEMM
// Out[nrows x 128] = A[nrows x 128] @ W[128 x 128] (+ bias per column)
// Block = 256 threads = 8 waves; each wave owns a 16-row x 128-col stripe.
// LDS holds W packed in K-pairs: sWp[(k>>1)*256 + col*2 + (k&1)] = W[k][col],
// so a B fragment (two K rows for one 16-col tile) is a single ds_load_b64.
__global__ __launch_bounds__(256)
void k_gemm128_wmma(const float* __restrict__ A, const float* __restrict__ W,
                    const float* __restrict__ bias, float* __restrict__ Out,
                    int nrows) {
    __shared__ float sWp[HID * HID];                      // 64 KB, K-pair packed
    const int t = threadIdx.x;
    for (int i = t * 4; i < HID * HID; i += 256 * 4) {
        float4 w4 = *reinterpret_cast<const float4*>(&W[i]);   // W[k][c..c+3]
        int k = i >> 7;
        int c = i & (HID - 1);
        float* dst0 = &sWp[(k >> 1) * 256 + (k & 1)];
        dst0[(c + 0) * 2] = w4.x;
        dst0[(c + 1) * 2] = w4.y;
        dst0[(c + 2) * 2] = w4.z;
        dst0[(c + 3) * 2] = w4.w;
    }
    __syncthreads();

    const int wave = t >> 5;
    const int lane = t & 31;
    const long row0 = ((long)blockIdx.x * 8 + wave) * 16;
    if (row0 >= nrows) return;                            // whole-wave exit; EXEC stays full

    const int lh   = lane & 15;                           // position within half-wave
    const int koff = (lane >> 4) << 1;                    // lanes 0-15 -> K+0,1 ; 16-31 -> K+2,3

    const float* arow = A + (row0 + lh) * (long)HID;

    // init accumulators with the per-column bias (same for every row)
    const float bv0 = bias ? bias[0 * 16 + lh] : 0.0f;
    v8f acc[8];
#pragma unroll
    for (int n2 = 0; n2 < 8; ++n2) {
        const float bv = bias ? bias[n2 * 16 + lh] : 0.0f;
        acc[n2] = (v8f)(bv);
    }
    (void)bv0;

    // software-pipelined A fragment: prefetch next K-step during WMMA chain
    v2f a_next = *reinterpret_cast<const v2f*>(arow + koff);
    for (int k = 0; k < HID; k += 4) {
        v2f a = a_next;
        if (k + 4 < HID)
            a_next = *reinterpret_cast<const v2f*>(arow + (k + 4) + koff);
        const float* bbase = &sWp[((k + koff) >> 1) * 256 + lh * 2];
#pragma unroll
        for (int n2 = 0; n2 < 8; ++n2) {
            v2f b = *reinterpret_cast<const v2f*>(bbase + n2 * 32);  // ds_load_b64
            acc[n2] = __builtin_amdgcn_wmma_f32_16x16x4_f32(
                false, a, false, b, (short)0, acc[n2], false, false);
        }
    }

    // C/D layout: VGPR j -> M = j (lanes 0-15) / M = j+8 (lanes 16-31), N = lh
    const long mbase = row0 + ((lane >= 16) ? 8 : 0);
#pragma unroll
    for (int n2 = 0; n2 < 8; ++n2) {
        const int col = n2 * 16 + lh;
#pragma unroll
        for (int j = 0; j < 8; ++j)
            Out[(mbase + j) * (long)HID + col] = acc[n2][j];
    }
}

// --------------------------------------------------- per-hop elementwise init
// agg = xw * self_norm[row] + b[col]   (scatter then adds on top of this)
__global__ void k_init_agg(const float* __restrict__ xw, const float* __restrict__ self_norm,
                           const float* __restrict__ bias, float* __restrict__ agg, int total) {
    int i = blockIdx.x * blockDim.x + threadIdx.x;
    if (i < total) {
        int row = i >> 7;
        int col = i & (HID - 1);
        agg[i] = xw[i] * self_norm[row] + bias[col];
    }
}

// ----------------------------------------------------------- edge scatter-add
// One wave per edge; each lane handles 4 consecutive features (float4 load,
// 4 global_atomic_add_f32 into the L2-resident agg buffer).
__global__ __launch_bounds__(256)
void k_scatter(const int* __restrict__ src, const int* __restrict__ dst,
               const float* __restrict__ en, const float* __restrict__ xw,
               float* __restrict__ agg, int e) {
    int t = blockIdx.x * blockDim.x + threadIdx.x;
    int edge = t >> 5;
    if (edge >= e) return;
    int lane = t & 31;
    int s = src[edge];
    int d = dst[edge];
    float w = en[edge];
    const float4 v = *reinterpret_cast<const float4*>(xw + (long)s * HID + lane * 4);
    float* o = agg + (long)d * HID + lane * 4;
    atomicAdd(o + 0, v.x * w);
    atomicAdd(o + 1, v.y * w);
    atomicAdd(o + 2, v.z * w);
    atomicAdd(o + 3, v.w * w);
}

__global__ void k_relu(float* p, int n) {
    int i = blockIdx.x * blockDim.x + threadIdx.x;
    if (i < n) p[i] = fmaxf(p[i], 0.0f);
}

// --------------------------------------------- classifier + graph pooling
// y[row] = h[row] @ W_cls + b_cls, atomically accumulated per graph.
__global__ __launch_bounds__(256)
void k_cls_pool(const float* __restrict__ h, const float* __restrict__ Wc,
                const float* __restrict__ bc, const int* __restrict__ batch,
                float* __restrict__ sums, float* __restrict__ cnts, int n) {
    __shared__ float sWc[HID * ODIM];
    __shared__ float sbc[ODIM];
    int t = threadIdx.x;
    for (int i = t; i < HID * ODIM; i += 256) sWc[i] = Wc[i];
    if (t < ODIM) sbc[t] = bc[t];
    __syncthreads();

    int row = blockIdx.x * 256 + t;
    if (row >= n) return;

    float acc[ODIM];
#pragma unroll
    for (int o = 0; o < ODIM; ++o) acc[o] = sbc[o];

    const float* hr = h + (long)row * HID;
    for (int k = 0; k < HID; ++k) {
        float hv = hr[k];
#pragma unroll
        for (int o = 0; o < ODIM; ++o) acc[o] += hv * sWc[k * ODIM + o];
    }
    int g = batch[row];
#pragma unroll
    for (int o = 0; o < ODIM; ++o) atomicAdd(&sums[g * ODIM + o], acc[o]);
    atomicAdd(&cnts[g], 1.0f);
}

__global__ void k_finalize(const float* __restrict__ sums, const float* __restrict__ cnts,
                           float* __restrict__ out, int total) {
    int i = blockIdx.x * blockDim.x + threadIdx.x;
    if (i < total) out[i] = sums[i] / fmaxf(cnts[i / ODIM], 1.0f);
}

// ---------------------------------------------------------------------------
extern "C" void kernel_launch(void* const* d_in, const int* in_sizes, int n_in,
                              void* d_out, int out_size, void* d_ws, size_t ws_size,
                              hipStream_t stream) {
    // inputs (setup_inputs dict order)
    const float* x      = (const float*)d_in[0];   // [N,128]
    const int*   eidx   = (const int*)  d_in[1];   // [2,E]
    const int*   batch  = (const int*)  d_in[2];   // [N]
    const float* W_in   = (const float*)d_in[3];   // [128,128]
    const float* b_in   = (const float*)d_in[4];   // [128]
    const float* W_hops = (const float*)d_in[5];   // [4,128,128]
    const float* b_hops = (const float*)d_in[6];   // [4,128]
    const float* W_cls  = (const float*)d_in[7];   // [128,10]
    const float* b_cls  = (const float*)d_in[8];   // [10]
    float*       out    = (float*)d_out;           // [G,10]

    const int N = in_sizes[0] / HID;
    const int E = in_sizes[1] / 2;
    const int G = out_size / ODIM;
    const int HOPS = in_sizes[6] / HID;            // b_hops = [HOPS,128]
    const int NH = N * HID;

    const int* src = eidx;
    const int* dst = eidx + E;

    // workspace layout (floats), 256B aligned
    auto align_up = [](size_t v) { return (v + 63) & ~(size_t)63; };
    float* p = (float*)d_ws;
    float* bufA      = p;                 p += align_up((size_t)NH);
    float* bufXW     = p;                 p += align_up((size_t)NH);
    float* bufB      = p;                 p += align_up((size_t)NH);
    float* dinv      = p;                 p += align_up((size_t)N);
    float* self_norm = p;                 p += align_up((size_t)N);
    float* edge_norm = p;                 p += align_up((size_t)E);
    float* pool_sums = p;                 p += align_up((size_t)(G * ODIM));
    float* pool_cnts = p;                 p += align_up((size_t)G);
    (void)ws_size;

    const int TB = 256;
    // ---- degrees / norms (reused all hops)
    k_fill<<<(N + TB - 1) / TB, TB, 0, stream>>>(dinv, 1.0f, N);             // deg = 1 (self loop)
    k_count_deg<<<(E + TB - 1) / TB, TB, 0, stream>>>(dst, dinv, E);         // deg += indeg
    k_dinv<<<(N + TB - 1) / TB, TB, 0, stream>>>(dinv, self_norm, N);        // dinv, dinv^2
    k_edge_norm<<<(E + TB - 1) / TB, TB, 0, stream>>>(src, dst, dinv, edge_norm, E);

    const int gemm_blocks = ((N + 15) / 16 + 7) / 8;

    // ---- encoder: h = x @ W_in + b_in
    k_gemm128_wmma<<<gemm_blocks, TB, 0, stream>>>(x, W_in, b_in, bufA, N);

    // ---- hops (ping-pong bufA <-> bufB)
    float* hcur = bufA;
    float* hnext = bufB;
    for (int hop = 0; hop < HOPS; ++hop) {
        k_gemm128_wmma<<<gemm_blocks, TB, 0, stream>>>(hcur, W_hops + (size_t)hop * HID * HID,
                                                       nullptr, bufXW, N);
        k_init_agg<<<(NH + TB - 1) / TB, TB, 0, stream>>>(bufXW, self_norm,
                                                          b_hops + (size_t)hop * HID, hnext, NH);
        k_scatter<<<(E * 32 + TB - 1) / TB, TB, 0, stream>>>(src, dst, edge_norm, bufXW, hnext, E);
        if (hop != HOPS - 1)
            k_relu<<<(NH + TB - 1) / TB, TB, 0, stream>>>(hnext, NH);
        float* tmp = hcur; hcur = hnext; hnext = tmp;
    }

    // ---- classifier + mean pool
    k_fill<<<(G * ODIM + TB - 1) / TB, TB, 0, stream>>>(pool_sums, 0.0f, G * ODIM);
    k_fill<<<(G + TB - 1) / TB, TB, 0, stream>>>(pool_cnts, 0.0f, G);
    k_cls_pool<<<(N + TB - 1) / TB, TB, 0, stream>>>(hcur, W_cls, b_cls, batch,
                                                     pool_sums, pool_cnts, N);
    k_finalize<<<(G * ODIM + TB - 1) / TB, TB, 0, stream>>>(pool_sums, pool_cnts, out, G * ODIM);
}